// ClusteringLayer_8005819040313
// MI455X (gfx1250) — compile-verified
//
#include <hip/hip_runtime.h>

typedef __attribute__((ext_vector_type(2))) float v2f;
typedef __attribute__((ext_vector_type(4))) float v4f;
typedef __attribute__((ext_vector_type(8))) float v8f;

#define DIM            256     // D
#define KC             64      // number of clusters
#define DSTEPS         (DIM / 4)
#define ROWS_PER_WAVE  16
#define WAVES_PER_BLK  8
#define ROWS_PER_BLK   (ROWS_PER_WAVE * WAVES_PER_BLK)   // 128

// q = (1 + d2/alpha)^(-(alpha+1)/2) with ALPHA == 1.0  ->  q = 1/(1+d2)

__global__ __launch_bounds__(256) void ClusteringLayer_kernel(
    const float* __restrict__ x,     // (N, 256) row-major
    const float* __restrict__ cl,    // (64, 256) row-major
    float* __restrict__ out)         // (N, 64)  row-major
{
    // Clusters staged in WMMA B-fragment order: [t (d-step)][j (k-tile)][lane]
    // Lane l of fragment (t,j) holds clusters[16*j + (l&15)][4*t + 2*(l>>4) .. +1]
    // (B is 4x16 K×N; lane = N, VGPR0 holds Kdim {0,2} split by lane half,
    //  VGPR1 holds Kdim {1,3} — mirror of the ISA 16x4 A-matrix table.)
    __shared__ v2f   sB[DSTEPS * 4 * 32];        // 64 KB
    __shared__ float sC2[KC];                    // cluster squared norms

    const int tid  = threadIdx.x;
    const int lane = tid & 31;
    const int wave = tid >> 5;

    // -------- stage clusters into LDS (consecutive threads -> consecutive
    // 8B LDS addresses: conflict-free ds_store_b64) --------
    for (int e = tid; e < DSTEPS * 4 * 32; e += 256) {
        const int t = e >> 7;           // d-step
        const int j = (e >> 5) & 3;     // k-tile
        const int l = e & 31;
        const int row = 16 * j + (l & 15);
        const int col = 4 * t + 2 * (l >> 4);
        sB[e] = *(const v2f*)(cl + row * DIM + col);
    }

    // -------- cluster norms once per block (identical for all waves; keep
    // this OUT of the WMMA loop so the matrix pipe isn't competing with
    // vector FMAs for issue slots) --------
    if (tid < KC) {
        const v4f* cp = (const v4f*)(cl + tid * DIM);
        float s0 = 0.f, s1 = 0.f;
#pragma unroll 8
        for (int d = 0; d < DIM / 4; ++d) {
            const v4f v = cp[d];
            s0 = fmaf(v.x, v.x, fmaf(v.y, v.y, s0));
            s1 = fmaf(v.z, v.z, fmaf(v.w, v.w, s1));
        }
        sC2[tid] = s0 + s1;
    }
    __syncthreads();

    const int n0   = blockIdx.x * ROWS_PER_BLK + wave * ROWS_PER_WAVE;
    const int arow = lane & 15;         // M index of this lane's A elements
    const int half = lane >> 4;         // 0: Kdim {0,1}, 1: Kdim {2,3}

    // A-fragment pointer: lane l streams row (n0 + l&15), columns 4t+2h, 4t+2h+1
    const float* aptr = x + (size_t)(n0 + arow) * DIM + 2 * half;

    v8f acc0 = {}, acc1 = {}, acc2 = {}, acc3 = {};
    float x2p = 0.f;                    // partial ||x_row||^2

#pragma unroll 4
    for (int t = 0; t < DSTEPS; ++t) {
        const v2f a = *(const v2f*)(aptr + 4 * t);
        x2p = fmaf(a.x, a.x, fmaf(a.y, a.y, x2p));

        const v2f b0 = sB[(t * 4 + 0) * 32 + lane];
        const v2f b1 = sB[(t * 4 + 1) * 32 + lane];
        const v2f b2 = sB[(t * 4 + 2) * 32 + lane];
        const v2f b3 = sB[(t * 4 + 3) * 32 + lane];

        // D = A*B + C, full f32 accumulation on the matrix pipe
        acc0 = __builtin_amdgcn_wmma_f32_16x16x4_f32(false, a, false, b0,
                                                     (short)0, acc0, false, false);
        acc1 = __builtin_amdgcn_wmma_f32_16x16x4_f32(false, a, false, b1,
                                                     (short)0, acc1, false, false);
        acc2 = __builtin_amdgcn_wmma_f32_16x16x4_f32(false, a, false, b2,
                                                     (short)0, acc2, false, false);
        acc3 = __builtin_amdgcn_wmma_f32_16x16x4_f32(false, a, false, b3,
                                                     (short)0, acc3, false, false);
    }

    // Lane l and l^16 covered complementary halves of the d-range.
    x2p += __shfl_xor(x2p, 16, 32);     // lane l now holds ||x_{n0 + (l&15)}||^2

    // Per-lane cluster norms for this lane's output columns (LDS broadcast read).
    const float c20 = sC2[ 0 + (lane & 15)];
    const float c21 = sC2[16 + (lane & 15)];
    const float c22 = sC2[32 + (lane & 15)];
    const float c23 = sC2[48 + (lane & 15)];

    // C/D layout: VGPR r, lanes 0-15 -> row r, lanes 16-31 -> row r+8; N = lane&15.
    float* obase = out + (size_t)n0 * KC + (lane & 15);

#pragma unroll
    for (int r = 0; r < 8; ++r) {
        const int row = r + 8 * half;                 // row within the 16-row tile
        const float x2r = __shfl(x2p, row, 32);       // broadcast that row's ||x||^2

        float d0 = fmaf(-2.f, acc0[r], x2r + c20);
        float d1 = fmaf(-2.f, acc1[r], x2r + c21);
        float d2 = fmaf(-2.f, acc2[r], x2r + c22);
        float d3 = fmaf(-2.f, acc3[r], x2r + c23);
        d0 = fmaxf(d0, 0.f); d1 = fmaxf(d1, 0.f);
        d2 = fmaxf(d2, 0.f); d3 = fmaxf(d3, 0.f);

        const float q0 = 1.f / (1.f + d0);
        const float q1 = 1.f / (1.f + d1);
        const float q2 = 1.f / (1.f + d2);
        const float q3 = 1.f / (1.f + d3);

        // Row sum over all 64 clusters: 4 per lane, butterfly over the 16-lane half.
        float p = (q0 + q1) + (q2 + q3);
        p += __shfl_xor(p, 1, 32);
        p += __shfl_xor(p, 2, 32);
        p += __shfl_xor(p, 4, 32);
        p += __shfl_xor(p, 8, 32);
        const float inv = 1.f / p;

        float* o = obase + (size_t)row * KC;
        o[0]  = q0 * inv;
        o[16] = q1 * inv;
        o[32] = q2 * inv;
        o[48] = q3 * inv;
    }
}

extern "C" void kernel_launch(void* const* d_in, const int* in_sizes, int n_in,
                              void* d_out, int out_size, void* d_ws, size_t ws_size,
                              hipStream_t stream) {
    const float* x  = (const float*)d_in[0];   // (N, 256) f32
    const float* cl = (const float*)d_in[1];   // (64, 256) f32
    float* out = (float*)d_out;                // (N, 64) f32

    const int n = in_sizes[0] / DIM;           // 524288 (multiple of 128)
    dim3 grid(n / ROWS_PER_BLK);
    dim3 block(256);
    ClusteringLayer_kernel<<<grid, block, 0, stream>>>(x, cl, out);
}